// GATLayer_86964497809686
// MI455X (gfx1250) — compile-verified
//
#include <hip/hip_runtime.h>
#include <hip/hip_bf16.h>
#include <math.h>

#define NN 50000           // nodes
#define NE 800000          // edges (without self loops)
#define ET (NE + NN)       // edges + self loops
#define IND 64             // input dim
#define H 4                // heads
#define D 64               // head dim
#define OD (H * D)         // 256 output dim
#define NEG_SLOPE 0.2f
#define EPS 1e-16f

typedef __attribute__((ext_vector_type(2))) float v2f;
typedef __attribute__((ext_vector_type(8))) float v8f;

// ---------------------------------------------------------------------------
// 0) init: out = bias (broadcast), m = -inf, denom = 0
// ---------------------------------------------------------------------------
__global__ __launch_bounds__(256) void k_init(float* __restrict__ out,
                                              const float* __restrict__ bias,
                                              float* __restrict__ m,
                                              float* __restrict__ denom) {
  long t = (long)blockIdx.x * blockDim.x + threadIdx.x;   // 0 .. NN*OD-1
  out[t] = bias[t & (OD - 1)];
  if (t < (long)NN * H) {
    ((unsigned*)m)[t] = 0xFF800000u;  // -inf
    denom[t] = 0.0f;
  }
}

// ---------------------------------------------------------------------------
// 1) h = x @ W via V_WMMA_F32_16X16X4_F32. One wave per 16x16 tile of h.
//    A (16x4 f32, 2 VGPRs): lanes 0-15 hold K={k0,k0+1}, lanes 16-31 K={k0+2,k0+3}
//    B (4x16 f32, 2 VGPRs): VGPR0 rows {k0, k0+2}, VGPR1 rows {k0+1, k0+3}
//    C/D (16x16, 8 VGPRs):  VGPR v -> (M = m0+v | m0+v+8, N = n0+lane15)
// ---------------------------------------------------------------------------
__global__ __launch_bounds__(256) void k_gemm(const float* __restrict__ x,
                                              const float* __restrict__ W,
                                              float* __restrict__ h) {
  const int lane = threadIdx.x & 31;
  const int wave = (blockIdx.x * (blockDim.x >> 5)) + (threadIdx.x >> 5);
  const int ntn  = OD / 16;                       // 16 tiles along N
  const int m0 = (wave / ntn) * 16;
  const int n0 = (wave % ntn) * 16;
  const int half = lane >> 4;                     // 0: lanes 0-15, 1: lanes 16-31
  const int l16  = lane & 15;

  const float* xrow = x + (long)(m0 + l16) * IND; // A row for this lane
  const float* wcol = W + n0 + l16;               // B column for this lane

  v8f acc = {};
#pragma unroll
  for (int k0 = 0; k0 < IND; k0 += 4) {
    const int kk = k0 + 2 * half;
    v2f a = *(const v2f*)(xrow + kk);             // K = kk, kk+1
    v2f b;
    b.x = wcol[(long)(kk + 0) * OD];              // row kk
    b.y = wcol[(long)(kk + 1) * OD];              // row kk+1
    acc = __builtin_amdgcn_wmma_f32_16x16x4_f32(false, a, false, b,
                                                (short)0, acc, false, false);
  }

  float* hp = h + (long)(m0 + half * 8) * OD + n0 + l16;
#pragma unroll
  for (int v = 0; v < 8; ++v) hp[(long)v * OD] = acc[v];
}

// ---------------------------------------------------------------------------
// 2) a_src[n,hd] = h[n,hd,:].att_src[hd,:], same for a_dst. Thread per (n,hd).
// ---------------------------------------------------------------------------
__global__ __launch_bounds__(256) void k_attn(const float* __restrict__ h,
                                              const float* __restrict__ att_src,
                                              const float* __restrict__ att_dst,
                                              float* __restrict__ a_src,
                                              float* __restrict__ a_dst) {
  long t = (long)blockIdx.x * blockDim.x + threadIdx.x;
  if (t >= (long)NN * H) return;
  const int hd = (int)(t & (H - 1));
  const long n = t >> 2;
  const float* hr = h + n * OD + hd * D;
  const float* as = att_src + hd * D;
  const float* ad = att_dst + hd * D;
  float s = 0.f, d = 0.f;
#pragma unroll 8
  for (int i = 0; i < D; ++i) {
    float v = hr[i];
    s += v * as[i];
    d += v * ad[i];
  }
  a_src[t] = s;
  a_dst[t] = d;
}

// sign-split float atomic max (m initialized to -inf bits)
__device__ __forceinline__ void atomicMaxF(float* addr, float val) {
  if (val >= 0.0f) atomicMax((int*)addr, __float_as_int(val));
  else             atomicMin((unsigned*)addr, __float_as_uint(val));
}

__device__ __forceinline__ void edge_sd(const int* __restrict__ ei, int e,
                                        int& s, int& d) {
  if (e < NE) { s = ei[e]; d = ei[NE + e]; }
  else        { s = d = e - NE; }
}

// ---------------------------------------------------------------------------
// 3) pass A: score = leaky_relu(a_src[src]+a_dst[dst]); segment max into m
// ---------------------------------------------------------------------------
__global__ __launch_bounds__(256) void k_score(const int* __restrict__ ei,
                                               const float* __restrict__ a_src,
                                               const float* __restrict__ a_dst,
                                               float* __restrict__ score,
                                               float* __restrict__ m) {
  long t = (long)blockIdx.x * blockDim.x + threadIdx.x;  // over ET*H
  if (t >= (long)ET * H) return;
  const int hd = (int)(t & (H - 1));
  const int e  = (int)(t >> 2);
  int s, d; edge_sd(ei, e, s, d);
  float sc = a_src[(long)s * H + hd] + a_dst[(long)d * H + hd];
  sc = (sc > 0.f) ? sc : sc * NEG_SLOPE;
  score[t] = sc;
  atomicMaxF(&m[(long)d * H + hd], sc);
}

// ---------------------------------------------------------------------------
// 4) pass B: e = exp(score - m[dst]); score <- e; segment sum into denom
// ---------------------------------------------------------------------------
__global__ __launch_bounds__(256) void k_expsum(const int* __restrict__ ei,
                                                float* __restrict__ score,
                                                const float* __restrict__ m,
                                                float* __restrict__ denom) {
  long t = (long)blockIdx.x * blockDim.x + threadIdx.x;  // over ET*H
  if (t >= (long)ET * H) return;
  const int hd = (int)(t & (H - 1));
  const int e  = (int)(t >> 2);
  int s, d; edge_sd(ei, e, s, d);
  float ev = expf(score[t] - m[(long)d * H + hd]);
  score[t] = ev;
  atomicAdd(&denom[(long)d * H + hd], ev);
}

// ---------------------------------------------------------------------------
// 5) pass C: out[dst] += h[src] * (e / (denom[dst]+eps)). One wave per edge,
//    8 floats per lane (head = lane>>3), alpha broadcast via shuffle.
// ---------------------------------------------------------------------------
__global__ __launch_bounds__(256) void k_agg(const int* __restrict__ ei,
                                             const float* __restrict__ h,
                                             const float* __restrict__ score,
                                             const float* __restrict__ denom,
                                             float* __restrict__ out) {
  const int lane = threadIdx.x & 31;
  const int wave = (blockIdx.x * (blockDim.x >> 5)) + (threadIdx.x >> 5);
  int s, d; edge_sd(ei, wave, s, d);

  float alpha = 0.f;
  if (lane < H)
    alpha = score[(long)wave * H + lane] / (denom[(long)d * H + lane] + EPS);
  const int head = lane >> 3;            // this lane's 8 floats live in head lane/8
  const float a = __shfl(alpha, head);

  const float4* hs = (const float4*)(h + (long)s * OD + lane * 8);
  float4 v0 = hs[0], v1 = hs[1];
  float* op = out + (long)d * OD + lane * 8;
  atomicAdd(op + 0, v0.x * a); atomicAdd(op + 1, v0.y * a);
  atomicAdd(op + 2, v0.z * a); atomicAdd(op + 3, v0.w * a);
  atomicAdd(op + 4, v1.x * a); atomicAdd(op + 5, v1.y * a);
  atomicAdd(op + 6, v1.z * a); atomicAdd(op + 7, v1.w * a);
}

// ---------------------------------------------------------------------------
extern "C" void kernel_launch(void* const* d_in, const int* in_sizes, int n_in,
                              void* d_out, int out_size, void* d_ws, size_t ws_size,
                              hipStream_t stream) {
  const float* x       = (const float*)d_in[0];
  const int*   ei      = (const int*)  d_in[1];   // [2, NE]
  const float* W       = (const float*)d_in[2];   // [64, 256]
  const float* att_src = (const float*)d_in[3];   // [4, 64]
  const float* att_dst = (const float*)d_in[4];
  const float* bias    = (const float*)d_in[5];   // [256]
  float* out = (float*)d_out;

  // workspace layout (bytes)
  char* ws = (char*)d_ws;
  float* h     = (float*)(ws);                                // NN*OD   = 51.2 MB
  float* score = (float*)(ws + (size_t)NN * OD * 4);          // ET*H    = 13.6 MB
  float* a_src = (float*)(ws + (size_t)NN * OD * 4 + (size_t)ET * H * 4);
  float* a_dst = a_src + (size_t)NN * H;
  float* m     = a_dst + (size_t)NN * H;
  float* denom = m     + (size_t)NN * H;

  // 0) init out/m/denom
  k_init<<<(NN * OD) / 256, 256, 0, stream>>>(out, bias, m, denom);

  // 1) h = x @ W  (50000 tiles, 8 waves per 256-thread block)
  k_gemm<<<(NN / 16) * (OD / 16) / 8, 256, 0, stream>>>(x, W, h);

  // 2) per-(node,head) attention logits
  k_attn<<<((long)NN * H + 255) / 256, 256, 0, stream>>>(h, att_src, att_dst,
                                                         a_src, a_dst);

  // 3) scores + segment max
  k_score<<<((long)ET * H + 255) / 256, 256, 0, stream>>>(ei, a_src, a_dst,
                                                          score, m);

  // 4) exp + segment sum
  k_expsum<<<((long)ET * H + 255) / 256, 256, 0, stream>>>(ei, score, m, denom);

  // 5) normalize + scatter-add aggregation (one wave per edge, 8 waves/block)
  k_agg<<<ET / 8, 256, 0, stream>>>(ei, h, score, denom, out);
}